// ARIMA_4767413698829
// MI455X (gfx1250) — compile-verified
//
#include <hip/hip_runtime.h>

// Problem constants (from reference): y (B,S,C) fp32, phi/theta 3-vectors fp32.
#define B_  64
#define S_  4096
#define C_  128
#define L_  128                      // chunk length in time
#define K_  32                       // chunks: K_*L_ == S_
#define NL_ (B_ * C_)                // 8192 independent (b,c) sequences
#define LANES_ 16                    // (b,c) lanes per workgroup (= WMMA N)
#define NTHREADS_ (LANES_ * K_)      // 512 threads = 16 waves

// LDS e_part layout: EP(k, t, x) = k*CHUNKSTRIDE_ + t*ROWSTRIDE_ + x
// ROWSTRIDE_=19: phase-3 half-wave offset 8*19 = 152 == 24 (mod 64)
//   -> 16-bank windows [b0,b0+15] and [b0+24,b0+39] are disjoint (no conflict)
// chunk skew +16: phase-1 two-chunks-per-wave offset 128*19+16 == 16 (mod 64)
//   -> the two 16-bank windows are disjoint (no conflict)
#define ROWSTRIDE_   19
#define CHUNKSTRIDE_ (L_ * ROWSTRIDE_ + 16)          // 2448 floats
#define EP_FLOATS_   (K_ * CHUNKSTRIDE_)             // 78336 floats
#define H_OFF_       EP_FLOATS_                      // H: 128*4 floats
#define CIN_OFF_     (H_OFF_ + L_ * 4)               // Cin: K*4*16 floats
#define LDS_FLOATS_  (CIN_OFF_ + K_ * 4 * LANES_)    // 80896 floats
#define LDS_BYTES_   (LDS_FLOATS_ * 4)               // 323584 B <= 320 KB

typedef float v2f __attribute__((ext_vector_type(2)));
typedef float v8f __attribute__((ext_vector_type(8)));

// ---------------------------------------------------------------------------
// Single fused kernel. Workgroup = (lane tile lt: 16 consecutive c of one b,
// all S). Phases:
//   0: impulse responses h_j[t] of the MA recurrence into LDS H[t*4+j]
//   1: per-chunk particular solution (zero incoming state) -> LDS EP
//   2: 3x3 state scan across chunks -> LDS Cin (WMMA B-slabs, row 3 = 0)
//   3: per-wave WMMA fixup  e = H(16x4) x Cin(4x16) + EP(16x16) -> out
// WMMA VGPR layouts per CDNA5 ISA 7.12.2:
//   A (16x4):  lanes 0-15 M=lane, K={0,1}; lanes 16-31 M=lane-16, K={2,3}
//   B (4x16):  vgpr r: lanes 0-15 K=r, lanes 16-31 K=r+2; N = lane&15
//   C/D (16x16): vgpr r: lanes 0-15 M=r, lanes 16-31 M=r+8; N = lane&15
// ---------------------------------------------------------------------------
__global__ void __launch_bounds__(NTHREADS_) arima_fused(
    const float* __restrict__ y, const float* __restrict__ phi,
    const float* __restrict__ theta, float* __restrict__ out) {
    extern __shared__ float lds[];
    float* EP  = lds;
    float* Hh  = lds + H_OFF_;
    float* CIN = lds + CIN_OFF_;

    const int tid   = threadIdx.x;
    const int lt    = blockIdx.x;            // lane tile 0..511
    const int bb    = lt >> 3;               // batch
    const int cbase = (lt & 7) * LANES_;     // channel base (16 | C_)

    const float t0 = theta[0], t1 = theta[1], t2 = theta[2];

    // ---- Phase 0: homogeneous impulse responses (j = 3 column stays 0) ----
    if (tid < 4) {
        const int j = tid;
        float e1 = (j == 0) ? 1.f : 0.f;
        float e2 = (j == 1) ? 1.f : 0.f;
        float e3 = (j == 2) ? 1.f : 0.f;
        for (int t = 0; t < L_; ++t) {
            float e = -(t0 * e1 + t1 * e2 + t2 * e3);
            Hh[t * 4 + j] = e;
            e3 = e2; e2 = e1; e1 = e;
        }
    }

    // ---- Phase 1: chunk particular solutions into LDS EP ----
    {
        const float p0 = phi[0], p1 = phi[1], p2 = phi[2];
        const int k = tid >> 4;              // chunk 0..31
        const int x = tid & 15;              // lane in tile
        const float* yp = y + ((size_t)bb * S_ + (size_t)k * L_) * C_ + cbase + x;
        float* ep = EP + k * CHUNKSTRIDE_ + x;
        float yl1 = 0.f, yl2 = 0.f, yl3 = 0.f;
        if (k > 0) { yl1 = yp[-C_]; yl2 = yp[-2 * C_]; yl3 = yp[-3 * C_]; }
        float e1 = 0.f, e2 = 0.f, e3 = 0.f;
#pragma unroll 4
        for (int t = 0; t < L_; ++t) {
            float yt = yp[t * C_];
            float z  = yt - p0 * yl1 - p1 * yl2 - p2 * yl3;  // off critical path
            float w  = z  - t1 * e2  - t2 * e3;              // off critical path
            float e  = w  - t0 * e1;                         // 1 FMA serial chain
            ep[t * ROWSTRIDE_] = e;
            yl3 = yl2; yl2 = yl1; yl1 = yt;
            e3 = e2; e2 = e1; e1 = e;
        }
    }
    __syncthreads();

    // ---- Phase 2: serial 3x3 state scan across chunks (16 threads) ----
    // Chunk-final particular states are EP rows L-1, L-2, L-3 (no extra array).
    if (tid < LANES_) {
        const int x = tid;
        const float M00 = Hh[(L_-1)*4+0], M01 = Hh[(L_-1)*4+1], M02 = Hh[(L_-1)*4+2];
        const float M10 = Hh[(L_-2)*4+0], M11 = Hh[(L_-2)*4+1], M12 = Hh[(L_-2)*4+2];
        const float M20 = Hh[(L_-3)*4+0], M21 = Hh[(L_-3)*4+1], M22 = Hh[(L_-3)*4+2];
        float c1 = 0.f, c2 = 0.f, c3 = 0.f;
        for (int k = 0; k < K_; ++k) {
            CIN[(k*4 + 0) * LANES_ + x] = c1;
            CIN[(k*4 + 1) * LANES_ + x] = c2;
            CIN[(k*4 + 2) * LANES_ + x] = c3;
            CIN[(k*4 + 3) * LANES_ + x] = 0.f;      // K=4 zero pad row
            const float* ep = EP + k * CHUNKSTRIDE_ + x;
            float s0 = ep[(L_-1)*ROWSTRIDE_] + M00*c1 + M01*c2 + M02*c3;
            float s1 = ep[(L_-2)*ROWSTRIDE_] + M10*c1 + M11*c2 + M12*c3;
            float s2 = ep[(L_-3)*ROWSTRIDE_] + M20*c1 + M21*c2 + M22*c3;
            c1 = s0; c2 = s1; c3 = s2;
        }
    }
    __syncthreads();

    // ---- Phase 3: WMMA fixup per wave (2 chunks x 8 t-tiles), store out ----
    {
        const int l    = tid & 31;
        const int w    = tid >> 5;           // wave 0..15 (uniform per wave)
        const int n    = l & 15;
        const int half = l >> 4;             // 0 | 1
        float* obase = out + (size_t)bb * S_ * C_ + cbase + n;
#pragma unroll
        for (int kk = 0; kk < 2; ++kk) {
            const int k = 2 * w + kk;        // uniform within the wave
            v2f bvec;                        // B: incoming states, K rows
            bvec.x = CIN[(k*4 + half*2 + 0) * LANES_ + n];
            bvec.y = CIN[(k*4 + half*2 + 1) * LANES_ + n];
            const float* epk = EP + k * CHUNKSTRIDE_ + n;
#pragma unroll
            for (int tt = 0; tt < L_ / 16; ++tt) {
                const int tl = tt * 16 + n;  // A row = local t
                v2f avec = *(const v2f*)(Hh + tl * 4 + half * 2);  // 8B aligned
                v8f cvec;
#pragma unroll
                for (int r = 0; r < 8; ++r)
                    cvec[r] = epk[(tt * 16 + half * 8 + r) * ROWSTRIDE_];
                v8f d = __builtin_amdgcn_wmma_f32_16x16x4_f32(
                    /*neg_a=*/false, avec, /*neg_b=*/false, bvec,
                    /*c_mod=*/(short)0, cvec, /*reuse_a=*/false, /*reuse_b=*/false);
                float* op = obase + (size_t)(k * L_ + tt * 16 + half * 8) * C_;
#pragma unroll
                for (int r = 0; r < 8; ++r) op[r * C_] = d[r];
            }
        }
    }
}

// ---------------------------------------------------------------------------
extern "C" void kernel_launch(void* const* d_in, const int* in_sizes, int n_in,
                              void* d_out, int out_size, void* d_ws, size_t ws_size,
                              hipStream_t stream) {
    const float* y     = (const float*)d_in[0];   // (B,S,C) fp32
    const float* phi   = (const float*)d_in[1];   // (3,)
    const float* theta = (const float*)d_in[2];   // (3,)
    float* out = (float*)d_out;                   // (B,S,C) fp32
    (void)d_ws; (void)ws_size; (void)in_sizes; (void)n_in; (void)out_size;

    // 512 workgroups x 512 threads; 323,584 B dynamic LDS per workgroup.
    arima_fused<<<NL_ / LANES_, NTHREADS_, LDS_BYTES_, stream>>>(y, phi, theta, out);
}